// GCN_encoder_82781199663862
// MI455X (gfx1250) — compile-verified
//
#include <hip/hip_runtime.h>
#include <hip/hip_bf16.h>

// CDNA5 / gfx1250 GCN encoder.
// Dense GEMMs use V_WMMA_F32_16X16X4_F32 (full f32 precision; workload is
// store-BW + transcendental bound, so fp32 WMMA is the right ceiling).
// Sparse aggregation uses global_atomic_add_f32 edge scatter.

typedef __attribute__((ext_vector_type(2))) float v2f;
typedef __attribute__((ext_vector_type(8))) float v8f;

#define WMMA_F32_4(a, b, c) \
  __builtin_amdgcn_wmma_f32_16x16x4_f32(false, (a), false, (b), (short)0, (c), false, false)

// ---------------------------------------------------------------- utilities
__global__ void zero_f32(float* __restrict__ p, int n) {
  int i = blockIdx.x * blockDim.x + threadIdx.x;
  const int stride = gridDim.x * blockDim.x;
  for (; i < n; i += stride) p[i] = 0.0f;
}

// ------------------------------------------------- GEMM1: h1 = x @ W1
// x:[N,512]  W1:[512,128]  h1:[N,128].  One wave per 16x16 tile.
__global__ __launch_bounds__(256) void gemm_x_w1(const float* __restrict__ X,
                                                 const float* __restrict__ W,
                                                 float* __restrict__ H) {
  const int lane = threadIdx.x;
  const int half = lane >> 4;       // 0: K=0/1 slice, 1: K=2/3 slice
  const int l    = lane & 15;
  const int m0   = blockIdx.x << 4;
  const int n0   = threadIdx.y << 4;          // 8 waves -> 8 column tiles
  const float* arow = X + (m0 + l) * 512 + 2 * half;
  const float* bcol = W + (2 * half) * 128 + n0 + l;
  v8f c = {};
#pragma unroll 4
  for (int k0 = 0; k0 < 512; k0 += 4) {
    v2f a, b;
    a.x = arow[k0];
    a.y = arow[k0 + 1];
    b.x = bcol[k0 * 128];
    b.y = bcol[k0 * 128 + 128];
    c = WMMA_F32_4(a, b, c);
  }
#pragma unroll
  for (int v = 0; v < 8; ++v)
    H[(m0 + v + 8 * half) * 128 + n0 + l] = c[v];
}

// ------------------------------------------------- GEMM2: h2 = relu(z1) @ W2
// z1:[N,128]  W2:[128,256]  h2:[N,256].  ReLU folded into the A-fragment load.
__global__ __launch_bounds__(256) void gemm_z1_w2(const float* __restrict__ Z1,
                                                  const float* __restrict__ W,
                                                  float* __restrict__ H) {
  const int lane = threadIdx.x;
  const int half = lane >> 4;
  const int l    = lane & 15;
  const int m0   = blockIdx.x << 4;
  const int n0   = (blockIdx.y * 8 + threadIdx.y) << 4;  // 16 column tiles
  const float* arow = Z1 + (m0 + l) * 128 + 2 * half;
  const float* bcol = W + (2 * half) * 256 + n0 + l;
  v8f c = {};
#pragma unroll 4
  for (int k0 = 0; k0 < 128; k0 += 4) {
    v2f a, b;
    a.x = fmaxf(arow[k0], 0.0f);
    a.y = fmaxf(arow[k0 + 1], 0.0f);
    b.x = bcol[k0 * 256];
    b.y = bcol[k0 * 256 + 256];
    c = WMMA_F32_4(a, b, c);
  }
#pragma unroll
  for (int v = 0; v < 8; ++v)
    H[(m0 + v + 8 * half) * 256 + n0 + l] = c[v];
}

// ------------------------------------------------- GEMM3: h3 = relu(z2) @ W3
// z2:[N,256]  W3:[256,20]  h3:[N,20].  N=20 < 32: column-guarded B/stores.
__global__ __launch_bounds__(64) void gemm_z2_w3(const float* __restrict__ Z2,
                                                 const float* __restrict__ W,
                                                 float* __restrict__ H) {
  const int lane = threadIdx.x;
  const int half = lane >> 4;
  const int l    = lane & 15;
  const int m0   = blockIdx.x << 4;
  const int n0   = threadIdx.y << 4;          // 2 column tiles (cols 0..31)
  const int n    = n0 + l;
  const bool ok  = (n < 20);
  const float* arow = Z2 + (m0 + l) * 256 + 2 * half;
  v8f c = {};
#pragma unroll 4
  for (int k0 = 0; k0 < 256; k0 += 4) {
    v2f a, b;
    a.x = fmaxf(arow[k0], 0.0f);
    a.y = fmaxf(arow[k0 + 1], 0.0f);
    const int kb = k0 + 2 * half;
    b.x = ok ? W[kb * 20 + n] : 0.0f;          // zero-pad cols 20..31
    b.y = ok ? W[(kb + 1) * 20 + n] : 0.0f;
    c = WMMA_F32_4(a, b, c);                   // full EXEC at the WMMA itself
  }
  if (ok) {
#pragma unroll
    for (int v = 0; v < 8; ++v)
      H[(m0 + v + 8 * half) * 20 + n] = c[v];
  }
}

// --------------------------------------------- SpMM: Out[r] += v * H[c] scatter
__global__ void spmm_scatter(const int* __restrict__ rows,
                             const int* __restrict__ cols,
                             const float* __restrict__ vals,
                             const float* __restrict__ H,
                             float* __restrict__ Out, int nE, int feat) {
  const int t = threadIdx.x;
  for (int e = blockIdx.x; e < nE; e += gridDim.x) {
    const int r = rows[e];
    const int c = cols[e];
    const float v = vals[e];
    if (t < feat) atomicAdd(Out + r * feat + t, v * H[c * feat + t]);
  }
}

// --------------------------------- head: z_adj = sigmoid(z @ z^T), z:[N,20]
// One wave per 16x16 tile; K=20 -> five fully-unrolled f32 WMMAs.
__global__ __launch_bounds__(256) void zzt_sigmoid(const float* __restrict__ Z,
                                                   float* __restrict__ Out) {
  const int lane = threadIdx.x;
  const int half = lane >> 4;
  const int l    = lane & 15;
  const int TI   = 625;                       // 10000 / 16
  const int t    = blockIdx.x * 8 + threadIdx.y;   // wave-uniform tile id
  if (t >= TI * TI) return;                   // wave-uniform exit: EXEC intact
  const int m0 = (t / TI) << 4;
  const int n0 = (t % TI) << 4;
  const float* za = Z + (m0 + l) * 20 + 2 * half;
  const float* zb = Z + (n0 + l) * 20 + 2 * half;   // B = z^T fragment
  v8f c = {};
#pragma unroll
  for (int k0 = 0; k0 < 20; k0 += 4) {
    v2f a, b;
    a.x = za[k0];
    a.y = za[k0 + 1];
    b.x = zb[k0];
    b.y = zb[k0 + 1];
    c = WMMA_F32_4(a, b, c);
  }
#pragma unroll
  for (int v = 0; v < 8; ++v) {
    const float s = 1.0f / (1.0f + __expf(-c[v]));
    Out[(m0 + v + 8 * half) * 10000 + n0 + l] = s;
  }
}

// ---------------------------------------------------------------- launcher
extern "C" void kernel_launch(void* const* d_in, const int* in_sizes, int n_in,
                              void* d_out, int out_size, void* d_ws, size_t ws_size,
                              hipStream_t stream) {
  const float* x    = (const float*)d_in[0];   // [10000, 512]
  const float* ev   = (const float*)d_in[1];   // [E]
  const float* W1   = (const float*)d_in[2];   // [512, 128]
  const float* W2   = (const float*)d_in[3];   // [128, 256]
  const float* W3   = (const float*)d_in[4];   // [256, 20]
  const int*   rows = (const int*)d_in[5];     // [E]
  const int*   cols = (const int*)d_in[6];     // [E]
  const int E = in_sizes[1];                   // 320000
  const int N = 10000;

  float* ws = (float*)d_ws;
  float* h1 = ws;                              // [N,128]  1,280,000
  float* z1 = h1 + N * 128;                    // [N,128]  1,280,000
  float* h2 = z1 + N * 128;                    // [N,256]  2,560,000
  float* z2 = h2 + N * 256;                    // [N,256]  2,560,000
  float* h3 = z2 + N * 256;                    // [N,20]     200,000

  float* z    = (float*)d_out;                 // [N,20]
  float* zadj = z + N * 20;                    // [N,N]

  const dim3 waveTile8(32, 8, 1);

  // Layer 1: h1 = x@W1 ; z1 = scatter(edges, h1)   (ReLU folded into GEMM2)
  gemm_x_w1<<<N / 16, waveTile8, 0, stream>>>(x, W1, h1);
  zero_f32<<<1024, 256, 0, stream>>>(z1, N * 128);
  spmm_scatter<<<20000, 128, 0, stream>>>(rows, cols, ev, h1, z1, E, 128);

  // Layer 2: h2 = relu(z1)@W2 ; z2 = scatter(edges, h2)
  gemm_z1_w2<<<dim3(N / 16, 2, 1), waveTile8, 0, stream>>>(z1, W2, h2);
  zero_f32<<<1024, 256, 0, stream>>>(z2, N * 256);
  spmm_scatter<<<20000, 256, 0, stream>>>(rows, cols, ev, h2, z2, E, 256);

  // Layer 3: h3 = relu(z2)@W3 ; z = scatter(edges, h3)  (no activation)
  gemm_z2_w3<<<N / 16, dim3(32, 2, 1), 0, stream>>>(z2, W3, h3);
  zero_f32<<<256, 256, 0, stream>>>(z, N * 20);
  spmm_scatter<<<20000, 32, 0, stream>>>(rows, cols, ev, h3, z, E, 20);

  // Head: z_adj = sigmoid(z @ z^T)   (400 MB stream-out: the real bottleneck)
  const int tiles = (N / 16) * (N / 16);       // 390,625 wave-tiles
  zzt_sigmoid<<<(tiles + 7) / 8, waveTile8, 0, stream>>>(z, zadj);
}